// DistortionParametersOptimizer_19215683682577
// MI455X (gfx1250) — compile-verified
//
#include <hip/hip_runtime.h>
#include <math.h>

// Problem constants from the reference
#define IMG_W 640.0f
#define IMG_H 480.0f
#define NPTS  2048
#define NL    4096
#define TPB   256
#define PPT   (NPTS / TPB)   // 8 points per thread
#define PAIRS (PPT / 2)      // 4 float4 (2-point) chunks per thread

#if defined(__gfx1250__) && __has_builtin(__builtin_amdgcn_global_load_async_to_lds_b128)
#define USE_ASYNC_LDS 1
#else
#define USE_ASYNC_LDS 0
#endif

typedef int v4i __attribute__((vector_size(16)));
typedef __attribute__((address_space(1))) v4i* g_v4i_ptr;
typedef __attribute__((address_space(3))) v4i* l_v4i_ptr;

__device__ __forceinline__ float wave32_reduce(float v) {
    // wave32 butterfly-ish reduce (result valid in lane 0)
    #pragma unroll
    for (int off = 16; off > 0; off >>= 1)
        v += __shfl_down(v, off, 32);
    return v;
}

__global__ __launch_bounds__(TPB)
void distortion_line_fit_kernel(const float* __restrict__ in,
                                const float* __restrict__ centerp,
                                const float* __restrict__ alphap,
                                float* __restrict__ out)
{
    const int l    = blockIdx.x;
    const int t    = threadIdx.x;
    const int lane = t & 31;
    const int wave = t >> 5;

    const float cx    = centerp[0];
    const float cy    = centerp[1];
    const float alpha = alphap[0];
    const float invW  = 1.0f / IMG_W;
    const float invH  = 1.0f / IMG_H;

    const float* base = in + (size_t)l * (size_t)(NPTS * 2);

    float ux[PPT], uy[PPT];
    float Sxx = 0.f, Sxy = 0.f, Syy = 0.f, Sx = 0.f, Sy = 0.f;

#if USE_ASYNC_LDS
    __shared__ float pts[NPTS * 2];   // 16 KB staging, DMA target
    // Async DMA global -> LDS: 4 x B128 per lane (each B128 = 2 points)
    #pragma unroll
    for (int j = 0; j < PAIRS; ++j) {
        const int m = j * TPB + t;    // pair index 0..1023
        g_v4i_ptr g = (g_v4i_ptr)(base + 4 * m);
        l_v4i_ptr s = (l_v4i_ptr)(&pts[4 * m]);
        __builtin_amdgcn_global_load_async_to_lds_b128(g, s, 0, 0);
    }
#if __has_builtin(__builtin_amdgcn_s_wait_asynccnt)
    __builtin_amdgcn_s_wait_asynccnt(0);
#else
    asm volatile("s_wait_asynccnt 0" ::: "memory");
#endif
    // Each thread reads back only the slots its own wave's DMA wrote -> no barrier needed.
#endif

    #pragma unroll
    for (int j = 0; j < PAIRS; ++j) {
        const int m = j * TPB + t;
        float4 p;
#if USE_ASYNC_LDS
        p = *(const float4*)(&pts[4 * m]);        // ds_load_b128
#else
        p = *(const float4*)(base + 4 * m);       // global_load_b128
#endif
        // point 2m
        {
            float sx = fmaf(p.x, invW, -cx);
            float sy = fmaf(p.y, invH, -cy);
            float r2 = fmaf(sx, sx, sy * sy);
            float f  = fmaf(alpha, r2, 1.0f);
            float u  = fmaf(f, sx, cx);
            float v  = fmaf(f, sy, cy);
            ux[2 * j] = u; uy[2 * j] = v;
            Sxx = fmaf(u, u, Sxx); Sxy = fmaf(u, v, Sxy); Syy = fmaf(v, v, Syy);
            Sx += u; Sy += v;
        }
        // point 2m+1
        {
            float sx = fmaf(p.z, invW, -cx);
            float sy = fmaf(p.w, invH, -cy);
            float r2 = fmaf(sx, sx, sy * sy);
            float f  = fmaf(alpha, r2, 1.0f);
            float u  = fmaf(f, sx, cx);
            float v  = fmaf(f, sy, cy);
            ux[2 * j + 1] = u; uy[2 * j + 1] = v;
            Sxx = fmaf(u, u, Sxx); Sxy = fmaf(u, v, Sxy); Syy = fmaf(v, v, Syy);
            Sx += u; Sy += v;
        }
    }

    // ---- block reduction of the 5 Gram sums (wave32 shuffle + LDS) ----
    __shared__ float red[8][5];
    __shared__ float bc[4];       // broadcast: s, c, rho*c, rho*s
    float part[5] = {Sxx, Sxy, Syy, Sx, Sy};
    #pragma unroll
    for (int k = 0; k < 5; ++k) part[k] = wave32_reduce(part[k]);
    if (lane == 0) {
        #pragma unroll
        for (int k = 0; k < 5; ++k) red[wave][k] = part[k];
    }
    __syncthreads();

    if (t == 0) {
        float S[5] = {0.f, 0.f, 0.f, 0.f, 0.f};
        #pragma unroll
        for (int w = 0; w < 8; ++w)
            #pragma unroll
            for (int k = 0; k < 5; ++k) S[k] += red[w][k];

        // M = A^T A, normalized by N for conditioning.  A = [x, y, -1]
        const float invN = 1.0f / (float)NPTS;
        float m00 = S[0] * invN, m01 = S[1] * invN, m11 = S[2] * invN;
        float m02 = -S[3] * invN, m12 = -S[4] * invN, m22 = 1.0f;

        // smallest eigenvalue (trigonometric method for symmetric 3x3)
        float q  = (m00 + m11 + m22) * (1.0f / 3.0f);
        float a00 = m00 - q, a11 = m11 - q, a22 = m22 - q;
        float p1 = m01 * m01 + m02 * m02 + m12 * m12;
        float p2 = a00 * a00 + a11 * a11 + a22 * a22 + 2.0f * p1;
        float p  = sqrtf(p2 * (1.0f / 6.0f));
        float ip = (p > 1e-20f) ? (1.0f / p) : 0.0f;
        float b00 = a00 * ip, b01 = m01 * ip, b02 = m02 * ip;
        float b11 = a11 * ip, b12 = m12 * ip, b22 = a22 * ip;
        float detB = b00 * (b11 * b22 - b12 * b12)
                   - b01 * (b01 * b22 - b12 * b02)
                   + b02 * (b01 * b12 - b11 * b02);
        float r = 0.5f * detB;
        r = fminf(1.0f, fmaxf(-1.0f, r));
        float phi  = acosf(r) * (1.0f / 3.0f);
        float lmin = q + 2.0f * p * cosf(phi + 2.0943951023931953f); // +2pi/3 -> smallest

        // eigenvector = max-norm cross product of rows of (M - lmin I)
        float A00 = m00 - lmin, A11 = m11 - lmin, A22 = m22 - lmin;
        float v0x = m01 * m12 - m02 * A11, v0y = m02 * m01 - A00 * m12, v0z = A00 * A11 - m01 * m01;
        float v1x = m01 * A22 - m02 * m12, v1y = m02 * m02 - A00 * A22, v1z = A00 * m12 - m01 * m02;
        float v2x = A11 * A22 - m12 * m12, v2y = m12 * m02 - m01 * A22, v2z = m01 * m12 - A11 * m02;
        float n0 = v0x * v0x + v0y * v0y + v0z * v0z;
        float n1 = v1x * v1x + v1y * v1y + v1z * v1z;
        float n2 = v2x * v2x + v2y * v2y + v2z * v2z;
        float vx = v0x, vy = v0y, vz = v0z, nb = n0;
        if (n1 > nb) { vx = v1x; vy = v1y; vz = v1z; nb = n1; }
        if (n2 > nb) { vx = v2x; vy = v2y; vz = v2z; }

        // z normalized by sqrt(z0^2+z1^2): s = z1, c = z0, rho = z2 (sign-invariant)
        float n01 = sqrtf(vx * vx + vy * vy);
        n01 = fmaxf(n01, 1e-30f);
        float inv01 = 1.0f / n01;
        float cs  = vx * inv01;
        float sn  = vy * inv01;
        float rho = vz * inv01;
        bc[0] = sn; bc[1] = cs; bc[2] = rho * cs; bc[3] = rho * sn;
    }
    __syncthreads();

    const float sn = bc[0], cs = bc[1], rc = bc[2], rs = bc[3];

    // ---- error pass over register-resident undistorted points ----
    float err = 0.f;
    #pragma unroll
    for (int i = 0; i < PPT; ++i) {
        float x = ux[i], y = uy[i];
        float b0 = x - rc;
        float b1 = y - rs;
        float beta = fmaf(sn, b0, -cs * b1);
        float px = fmaf(beta, sn, rc);
        float py = fmaf(-beta, cs, rs);
        float dx = px - x, dy = py - y;
        err = fmaf(dx, dx, fmaf(dy, dy, err));
    }
    err = wave32_reduce(err);

    __shared__ float red2[8];
    if (lane == 0) red2[wave] = err;
    __syncthreads();
    if (t == 0) {
        float e = 0.f;
        #pragma unroll
        for (int w = 0; w < 8; ++w) e += red2[w];
        out[l] = e;
    }
}

extern "C" void kernel_launch(void* const* d_in, const int* in_sizes, int n_in,
                              void* d_out, int out_size, void* d_ws, size_t ws_size,
                              hipStream_t stream) {
    (void)in_sizes; (void)n_in; (void)d_ws; (void)ws_size; (void)out_size;
    const float* in     = (const float*)d_in[0];
    const float* center = (const float*)d_in[1];
    const float* alpha  = (const float*)d_in[2];
    float* out = (float*)d_out;
    distortion_line_fit_kernel<<<NL, TPB, 0, stream>>>(in, center, alpha, out);
}